// KVEmbedding_39187281609184
// MI455X (gfx1250) — compile-verified
//
#include <hip/hip_runtime.h>
#include <stdint.h>

// ---------------------------------------------------------------------------
// Embedding gather: out[n, 0:64] = table[idx[n], 0:64]   (n = 0 .. 3,276,799)
//
// Memory-roofline kernel (~72us floor: ~1.7GB at 23.3 TB/s, zero FLOPs).
// Uses the CDNA5 Tensor Data Mover GATHER MODE (purpose-built for embedding
// gathers): one TENSOR_LOAD_TO_LDS pulls 8 gathered rows (2KB) via 32-bit row
// indices in descriptor groups 2/3; one TENSOR_STORE_FROM_LDS streams the
// contiguous 2KB block out. Payload never touches VGPRs; waves only build
// descriptors in SGPRs (indices fetched with one s_load_b256).
//
// Pipeline: TENSORcnt ops of a wave retire strictly IN ORDER (loads+stores
// together), so with 4 LDS buffers we keep 2 loads + 2 stores (8KB) in flight
// per wave with a single s_wait_tensorcnt per 2KB group:
//   issue order  L0, L1, [S0,L2], [S1,L3], [S2,L4], ...
//   before S_i   : wait tc <= (i>=1) + (L_{i+1} exists)  ==> L_i retired
//   before L_{i+2}: S_{i-2} (its buffer's reader) is older than L_i, so the
//                   same wait already retired it -- no extra fence.
// 16384 waves x 8KB in flight ≈ 128MB outstanding DMA: plenty to cover HBM
// latency at 23.3 TB/s.
// ---------------------------------------------------------------------------

typedef __attribute__((ext_vector_type(4))) unsigned int v4u;
typedef __attribute__((ext_vector_type(8))) int          v8i;
typedef __attribute__((ext_vector_type(4))) int          v4i;

#define ROWS_PER_GROUP   8            // 32-bit-index gather mode: up to 8 rows
#define ROW_ELEMS        64           // EMBED_DIM
#define GROUP_BYTES      2048         // 8 rows * 64 f32 * 4B
#define NBUF             4            // LDS buffers per wave (2 loads + 2 stores in flight)
#define WAVES_PER_BLOCK  8            // 256 threads
#define VOCAB_ROWS       1000000

// Issue a gather-mode tensor load: writes 8 gathered rows to the 2KB of LDS at lds_addr.
__device__ __forceinline__ void tdm_gather_load(unsigned long long table_addr,
                                                unsigned int lds_addr,
                                                const int* __restrict__ ip) {
    v4u g0;
    // count=1 | gather_index_size=32b (bit30) | gather_mode=1 (bit31)
    g0.x = 0xC0000001u;
    g0.y = lds_addr;                                   // lds_addr[63:32 of D#]
    g0.z = (unsigned int)table_addr;                   // global_addr lo
    g0.w = (unsigned int)((table_addr >> 32) & 0x01FFFFFFu) | (2u << 30); // +type=2

    v8i g1;
    g1[0] = 0x00020000;                 // workgroup_mask=0, data_size=2 (4B)
    g1[1] = (ROW_ELEMS << 16);          // tensor_dim0 = 64 (lo16 at bits 63:48)
    g1[2] = (int)((VOCAB_ROWS & 0xFFFF) << 16);     // tensor_dim1 lo16 (1e6=0xF4240)
    g1[3] = (VOCAB_ROWS >> 16) | (ROW_ELEMS << 16); // tensor_dim1 hi16 | tile_dim0=64
    g1[4] = ROWS_PER_GROUP;             // tile_dim1 = #valid gather indices = 8
    g1[5] = ROW_ELEMS;                  // tensor_dim0_stride = 64 elements (lo32)
    g1[6] = 0;
    g1[7] = 0;

    v4i g2, g3;                         // gather mode: 32-bit row indices 0..7
    g2.x = ip[0]; g2.y = ip[1]; g2.z = ip[2]; g2.w = ip[3];
    g3.x = ip[4]; g3.y = ip[5]; g3.z = ip[6]; g3.w = ip[7];

    v8i gx;                             // extra operand of 6-arg builtin: zero
    gx[0] = 0; gx[1] = 0; gx[2] = 0; gx[3] = 0;
    gx[4] = 0; gx[5] = 0; gx[6] = 0; gx[7] = 0;

    __builtin_amdgcn_tensor_load_to_lds(g0, g1, g2, g3, gx, 0);
}

// Issue a normal-mode tensor store: copies 2KB from LDS at lds_addr to global memory at dst_addr.
__device__ __forceinline__ void tdm_store(unsigned long long dst_addr,
                                          unsigned int lds_addr) {
    v4u g0;
    g0.x = 0x00000001u;                                // count=1, normal mode
    g0.y = lds_addr;
    g0.z = (unsigned int)dst_addr;
    g0.w = (unsigned int)((dst_addr >> 32) & 0x01FFFFFFu) | (2u << 30);

    v8i g1;
    g1[0] = 0x00020000;                 // data_size = 4B
    g1[1] = (512 << 16);                // tensor_dim0 = 512 elements (2KB)
    g1[2] = (1 << 16);                  // tensor_dim1 = 1
    g1[3] = (512 << 16);                // tile_dim0 = 512
    g1[4] = 1;                          // tile_dim1 = 1
    g1[5] = 512;                        // tensor_dim0_stride = 512
    g1[6] = 0;
    g1[7] = 0;

    v4i gz; gz.x = 0; gz.y = 0; gz.z = 0; gz.w = 0;    // 2D tensor: groups 2/3 unused
    v8i gx;
    gx[0] = 0; gx[1] = 0; gx[2] = 0; gx[3] = 0;
    gx[4] = 0; gx[5] = 0; gx[6] = 0; gx[7] = 0;

    __builtin_amdgcn_tensor_store_from_lds(g0, g1, gz, gz, gx, 0);
}

__global__ void __launch_bounds__(WAVES_PER_BLOCK * 32)
kv_embedding_tdm_gather(const int* __restrict__ idx,      // [B*H] int32 row ids
                        const float* __restrict__ table,  // [1e6, 64]
                        float* __restrict__ out,          // [B*H, 64]
                        int n_groups) {                   // (B*H)/8
    __shared__ char smem[WAVES_PER_BLOCK * NBUF * GROUP_BYTES];

    // Wave-uniform identity (readfirstlane => everything downstream lives in
    // SGPRs: index loads become s_load_b256, descriptors build in SALU).
    const int wave     = __builtin_amdgcn_readfirstlane((int)(threadIdx.x >> 5));
    const int wave_gid = __builtin_amdgcn_readfirstlane((int)blockIdx.x) * WAVES_PER_BLOCK + wave;
    const int stride   = (int)gridDim.x * WAVES_PER_BLOCK;

    const unsigned long long table_addr = (unsigned long long)(uintptr_t)table;
    const unsigned long long out_addr   = (unsigned long long)(uintptr_t)out;
    // Generic->LDS: low 32 bits of a generic shared-memory address are the
    // workgroup-relative LDS byte offset (ISA aperture rules).
    const unsigned int lds0 =
        (unsigned int)(uintptr_t)(void*)&smem[wave * NBUF * GROUP_BYTES];

    int p = wave_gid;
    if (p >= n_groups) return;                 // wave-uniform exit

    // ---- prologue: issue up to 2 loads (buffers 0, 1) ----
    tdm_gather_load(table_addr, lds0, idx + (size_t)p * ROWS_PER_GROUP);
    if (p + stride < n_groups)
        tdm_gather_load(table_addr, lds0 + GROUP_BYTES,
                        idx + (size_t)(p + stride) * ROWS_PER_GROUP);

    unsigned int it = 0;
    for (;;) {
        const int pnext = p + stride;          // its load is already in flight
        const int p2    = p + 2 * stride;      // load issued this iteration
        const bool has_next  = (pnext < n_groups);

        // Retire L_it (and, in order, everything older -- incl. S_{it-2},
        // whose buffer L_{it+2} will overwrite).  Younger ops that may remain:
        // S_{it-1} (if it>=1) and L_{it+1} (if has_next).
        if (it == 0) {
            if (has_next) __builtin_amdgcn_s_wait_tensorcnt((short)1);
            else          __builtin_amdgcn_s_wait_tensorcnt((short)0);
        } else {
            if (has_next) __builtin_amdgcn_s_wait_tensorcnt((short)2);
            else          __builtin_amdgcn_s_wait_tensorcnt((short)1);
        }

        tdm_store(out_addr + (unsigned long long)p * GROUP_BYTES,
                  lds0 + (it & (NBUF - 1u)) * GROUP_BYTES);

        if (p2 < n_groups)
            tdm_gather_load(table_addr,
                            lds0 + ((it + 2u) & (NBUF - 1u)) * GROUP_BYTES,
                            idx + (size_t)p2 * ROWS_PER_GROUP);

        if (!has_next) break;                  // s_endpgm implies wait-idle
        p = pnext;
        ++it;
    }
}

extern "C" void kernel_launch(void* const* d_in, const int* in_sizes, int n_in,
                              void* d_out, int out_size, void* d_ws, size_t ws_size,
                              hipStream_t stream) {
    (void)n_in; (void)d_ws; (void)ws_size; (void)out_size;

    const int*   idx   = (const int*)d_in[0];     // indices (int32 per harness)
    const float* table = (const float*)d_in[1];   // [1e6, 64] f32
    float*       out   = (float*)d_out;           // [B*H, 64] f32
    // d_in[2] is the dummy scalar; reference adds dummy.sum()*0.0 == 0 -> ignore.

    const int n_rows   = in_sizes[0];             // B*H = 3,276,800
    const int n_groups = n_rows / ROWS_PER_GROUP; // 409,600 (exact)

    const int blocks = 2048;                      // 16384 waves, ~25 groups each
    kv_embedding_tdm_gather<<<blocks, WAVES_PER_BLOCK * 32, 0, stream>>>(
        idx, table, out, n_groups);
}